// SDPQuantizer_60266981097802
// MI455X (gfx1250) — compile-verified
//
#include <hip/hip_runtime.h>
#include <cstdint>
#include <cstddef>

typedef float v4f __attribute__((ext_vector_type(4)));
typedef int sdp_v4i __attribute__((vector_size(16)));  // matches builtin's V4i

#if __has_builtin(__builtin_amdgcn_global_load_async_to_lds_b128)
#define SDP_ASYNC 1
#else
#define SDP_ASYNC 0
#endif

#define SDP_AS1 __attribute__((address_space(1)))
#define SDP_AS3 __attribute__((address_space(3)))

// ---- order-preserving float<->uint encoding (for atomic min/max on floats) ----
__device__ __forceinline__ unsigned sdp_enc(float f) {
  unsigned u = __float_as_uint(f);
  return (u & 0x80000000u) ? ~u : (u | 0x80000000u);
}
__device__ __forceinline__ float sdp_dec(unsigned u) {
  unsigned b = (u & 0x80000000u) ? (u & 0x7fffffffu) : ~u;
  return __uint_as_float(b);
}

template <int N>
__device__ __forceinline__ void sdp_wait_async() {
#if __has_builtin(__builtin_amdgcn_s_wait_asynccnt)
  __builtin_amdgcn_s_wait_asynccnt(N);
#else
  asm volatile("s_wait_asynccnt %0" ::"i"(N) : "memory");
#endif
  asm volatile("" ::: "memory");  // keep LDS reads below the wait
}

// ---- per-group math: quantize 8 values, 4th-largest threshold, SDP dequant ----
__device__ __forceinline__ void sdp_process_group(const float v[8], float scale,
                                                  float o[8]) {
  float q[8], m[8];
#pragma unroll
  for (int j = 0; j < 8; ++j) {
    float t = rintf(v[j] / scale);           // RNE, matches jnp.round
    t = fminf(fmaxf(t, -128.0f), 127.0f);    // clip to [Q_MIN, Q_MAX]
    q[j] = t;
    m[j] = fabsf(t);
  }
  // Batcher odd-even mergesort network for 8 (ascending); we need index 4
  float a0 = m[0], a1 = m[1], a2 = m[2], a3 = m[3];
  float a4 = m[4], a5 = m[5], a6 = m[6], a7 = m[7];
#define SDP_CE(x, y)                        \
  {                                         \
    float lo_ = fminf(x, y), hi_ = fmaxf(x, y); \
    x = lo_;                                \
    y = hi_;                                \
  }
  SDP_CE(a0, a1) SDP_CE(a2, a3) SDP_CE(a4, a5) SDP_CE(a6, a7)
  SDP_CE(a0, a2) SDP_CE(a1, a3) SDP_CE(a4, a6) SDP_CE(a5, a7)
  SDP_CE(a1, a2) SDP_CE(a5, a6)
  SDP_CE(a0, a4) SDP_CE(a1, a5) SDP_CE(a2, a6) SDP_CE(a3, a7)
  SDP_CE(a2, a4) SDP_CE(a3, a5)
  SDP_CE(a1, a2) SDP_CE(a3, a4) SDP_CE(a5, a6)
#undef SDP_CE
  const float th = a4;  // 4th-largest (== top_k[..., -1]); ties pass via >=
#pragma unroll
  for (int j = 0; j < 8; ++j) {
    float mag = m[j];
    float high = floorf(mag * 0.0625f);      // mag / 16, exact (power of 2)
    float low = mag - high * 16.0f;
    float keep = (mag >= th) ? low : 0.0f;
    float val = high * 16.0f + keep;
    o[j] = copysignf(scale * val, q[j]);     // sign(x_int) * ... (0 stays 0)
  }
}

// ---- hot kernel first in .text so the disasm snippet shows it ----
// Streaming SDP quantize; reversed tiles for L2 reuse of pass-1 lines.
__global__ __launch_bounds__(256) void sdp_quant(const float* __restrict__ x,
                                                 float* __restrict__ out,
                                                 const unsigned* __restrict__ ws,
                                                 long long ngroups) {
  const float r_min = sdp_dec(ws[0]);
  const float r_max = sdp_dec(ws[1]);
  const float scale = fmaxf((r_max - r_min) / 255.0f, 1e-8f);

  const long long tiles = (ngroups + 255) / 256;
  const int tid = threadIdx.x;
  const long long tstride = gridDim.x;
  const long long t0 = tiles - 1 - (long long)blockIdx.x;  // reversed order

#if SDP_ASYNC
  // Double-buffered async global->LDS staging (ASYNCcnt pipeline).
  // Each lane copies its own 32B group into its own LDS slot; no cross-lane
  // sharing, so per-wave s_wait_asynccnt is the only sync needed.
  __shared__ float tile[2][256 * 8];  // 2 x 8KB

  auto issue = [&](long long t, int buf) {
    if (t < 0) return;
    long long g = t * 256 + tid;
    if (g < ngroups) {
      const float* src = x + g * 8;
      float* dst = &tile[buf][tid * 8];
      __builtin_amdgcn_global_load_async_to_lds_b128(
          (SDP_AS1 sdp_v4i*)src, (SDP_AS3 sdp_v4i*)dst, 0, 0);
      __builtin_amdgcn_global_load_async_to_lds_b128(
          (SDP_AS1 sdp_v4i*)(src + 4), (SDP_AS3 sdp_v4i*)(dst + 4), 0, 0);
    }
  };

  int buf = 0;
  issue(t0, 0);
  for (long long t = t0; t >= 0; t -= tstride) {
    const long long tn = t - tstride;
    if (tn >= 0) {
      issue(tn, buf ^ 1);     // up to 4 async copies in flight
      sdp_wait_async<2>();    // current tile's 2 copies complete (in-order)
    } else {
      sdp_wait_async<0>();
    }
    long long g = t * 256 + tid;
    if (g < ngroups) {
      const float* lp = &tile[buf][tid * 8];
      v4f a = *(const v4f*)lp;        // ds_load_b128
      v4f b = *(const v4f*)(lp + 4);
      float v[8] = {a.x, a.y, a.z, a.w, b.x, b.y, b.z, b.w};
      float o[8];
      sdp_process_group(v, scale, o);
      v4f oa = {o[0], o[1], o[2], o[3]};
      v4f ob = {o[4], o[5], o[6], o[7]};
      float* op = out + g * 8;
      __builtin_nontemporal_store(oa, (v4f*)op);       // streaming store
      __builtin_nontemporal_store(ob, (v4f*)(op + 4));
    }
    buf ^= 1;
  }
#else
  // Fallback: direct coalesced b128 loads (wave covers 1KB contiguous/load).
  for (long long t = t0; t >= 0; t -= tstride) {
    long long g = t * 256 + tid;
    if (g >= ngroups) continue;
    const float* src = x + g * 8;
    long long gp = (t - tstride) * 256 + tid;
    if (gp >= 0) __builtin_prefetch((const void*)(x + gp * 8), 0, 0);
    v4f a = *(const v4f*)src;
    v4f b = *(const v4f*)(src + 4);
    float v[8] = {a.x, a.y, a.z, a.w, b.x, b.y, b.z, b.w};
    float o[8];
    sdp_process_group(v, scale, o);
    v4f oa = {o[0], o[1], o[2], o[3]};
    v4f ob = {o[4], o[5], o[6], o[7]};
    float* op = out + g * 8;
    __builtin_nontemporal_store(oa, (v4f*)op);
    __builtin_nontemporal_store(ob, (v4f*)(op + 4));
  }
#endif
}

// ---- kernel 0: reset ws accumulators (min=enc(0), max=enc(0)) every call ----
__global__ void sdp_init_ws(unsigned* ws) {
  ws[0] = 0x80000000u;  // enc(+0.0f): r_min starts at 0
  ws[1] = 0x80000000u;  // enc(+0.0f): r_max starts at 0
}

// ---- kernel 1: global min/max (bandwidth bound; keep lines resident in L2) ----
__global__ __launch_bounds__(256) void sdp_minmax(const float* __restrict__ x,
                                                  unsigned* __restrict__ ws,
                                                  long long n4) {
  const v4f* xv = (const v4f*)x;
  float vmin = 0.0f, vmax = 0.0f;
  long long i = (long long)blockIdx.x * 256ll + threadIdx.x;
  const long long stride = (long long)gridDim.x * 256ll;
  for (; i < n4; i += stride) {
    long long ipf = i + 8 * stride;  // deep prefetch -> global_prefetch_b8
    if (ipf < n4) __builtin_prefetch((const void*)(xv + ipf), 0, 0);
    v4f v = xv[i];  // global_load_b128
    vmin = fminf(vmin, fminf(fminf(v.x, v.y), fminf(v.z, v.w)));
    vmax = fmaxf(vmax, fmaxf(fmaxf(v.x, v.y), fmaxf(v.z, v.w)));
  }
  __shared__ float smin[256];
  __shared__ float smax[256];
  const int tid = threadIdx.x;
  smin[tid] = vmin;
  smax[tid] = vmax;
  __syncthreads();
  for (int s = 128; s > 0; s >>= 1) {
    if (tid < s) {
      smin[tid] = fminf(smin[tid], smin[tid + s]);
      smax[tid] = fmaxf(smax[tid], smax[tid + s]);
    }
    __syncthreads();
  }
  if (tid == 0) {
    atomicMin(&ws[0], sdp_enc(smin[0]));
    atomicMax(&ws[1], sdp_enc(smax[0]));
  }
}

extern "C" void kernel_launch(void* const* d_in, const int* in_sizes, int n_in,
                              void* d_out, int out_size, void* d_ws,
                              size_t ws_size, hipStream_t stream) {
  const float* x = (const float*)d_in[0];
  float* out = (float*)d_out;
  unsigned* ws = (unsigned*)d_ws;

  const long long n = (long long)in_sizes[0];  // 8*4096*2048 = 67,108,864
  const long long n4 = n / 4;
  const long long ngroups = n / 8;

  sdp_init_ws<<<1, 1, 0, stream>>>(ws);

  const int mm_blocks = 4096;  // 1M threads, ~16 float4 each (grid-stride)
  sdp_minmax<<<mm_blocks, 256, 0, stream>>>(x, ws, n4);

  const long long tiles = (ngroups + 255) / 256;
  const int q_blocks = (int)((tiles < 4096) ? tiles : 4096);
  sdp_quant<<<q_blocks, 256, 0, stream>>>(x, out, ws, ngroups);
}